// VARGLLVM_32555852103915
// MI455X (gfx1250) — compile-verified
//
#include <hip/hip_runtime.h>

typedef __attribute__((ext_vector_type(2))) float v2f;
typedef __attribute__((ext_vector_type(8))) float v8f;

#define B_    4096
#define T_    200
#define L_    8
#define V_    64
#define C_    16
#define NBERN 32
#define CLIPV 20.0f

// ---------------------------------------------------------------------------
// Kernel 0: fuse per-(b,v) constant term: uterm = intercepts + u*exp(.5*lvu).
// 1 MB table, L2-resident; becomes the WMMA C-operand seed in the GEMM.
// ---------------------------------------------------------------------------
__global__ __launch_bounds__(256) void uterm_kernel(
    const float* __restrict__ u, const float* __restrict__ intercepts,
    const float* __restrict__ logvar_u, float* __restrict__ uterm)
{
    int i = blockIdx.x * blockDim.x + threadIdx.x;
    if (i < B_ * V_) {
        int v = i & (V_ - 1);
        uterm[i] = fmaf(u[i], __expf(0.5f * logvar_u[v]), intercepts[v]);
    }
}

// ---------------------------------------------------------------------------
// Kernel 1: VAR(1) recursion. One thread per batch row; z[8] lives in VGPRs,
// transition matrix + betas in LDS. Writes z as (B, T, L) row-major so the
// GEMM kernel can fetch WMMA A-operands with single 64-bit loads.
// ---------------------------------------------------------------------------
__global__ __launch_bounds__(128) void var1_recursion_kernel(
    const float* __restrict__ eps, const float* __restrict__ A,
    const float* __restrict__ logvar_z1, const float* __restrict__ beta0,
    const float* __restrict__ beta1, float* __restrict__ z_out)
{
    __shared__ float sA[L_ * L_];
    __shared__ float sB0[L_], sB1[L_], sS[L_];
    int tid = threadIdx.x;
    if (tid < L_ * L_) sA[tid] = A[tid];
    if (tid < L_) {
        sB0[tid] = beta0[tid];
        sB1[tid] = beta1[tid];
        sS[tid]  = __expf(0.5f * logvar_z1[tid]);
    }
    __syncthreads();

    int b = blockIdx.x * blockDim.x + tid;
    if (b >= B_) return;

    const float* ep = eps   + (size_t)b * T_ * L_;
    float*       zp = z_out + (size_t)b * T_ * L_;

    float z[L_];
#pragma unroll
    for (int l = 0; l < L_; ++l) { z[l] = ep[l] * sS[l]; zp[l] = z[l]; }

    for (int t = 1; t < T_; ++t) {
        const float* e = ep + t * L_;
        float tt = (float)t;
        float zn[L_];
#pragma unroll
        for (int l = 0; l < L_; ++l) {
            float acc = fmaf(sB1[l], tt, sB0[l]) + e[l];
#pragma unroll
            for (int k = 0; k < L_; ++k)
                acc = fmaf(z[k], sA[k * L_ + l], acc);
            zn[l] = acc;
        }
#pragma unroll
        for (int l = 0; l < L_; ++l) { z[l] = zn[l]; zp[t * L_ + l] = z[l]; }
    }
}

// ---------------------------------------------------------------------------
// Kernel 2: fused [x|z] @ [wx;wz] + uterm, clip, link epilogue.
// Each wave computes a 16-row (flattened b*T+t) x 64-col strip with
// V_WMMA_F32_16X16X4_F32. The per-row u-term is fed in as the C-operand of
// the first WMMA of each chain (it is already in C/D layout); the clip is a
// single v_med3 via the fmed3 builtin. Strip id is wave-uniform; bases are
// forced to SALU via readfirstlane; EXEC stays all-ones for WMMA.
// ---------------------------------------------------------------------------
__global__ __launch_bounds__(256) void glm_wmma_kernel(
    const float* __restrict__ x,  const float* __restrict__ z,
    const float* __restrict__ uterm, const float* __restrict__ wx,
    const float* __restrict__ wz, float* __restrict__ out)
{
    const int lane = threadIdx.x & 31;
    const int half = lane >> 4;   // 0: lanes 0-15, 1: lanes 16-31
    const int n    = lane & 15;

    // --- B-operands (4x16 K-chunks). VGPR j, half h -> K = base + j + 2h. ---
    v2f bx[4][4];   // [n-tile][k-chunk] of wx (16 x 64)
    v2f bz[4][2];   // [n-tile][k-chunk] of wz ( 8 x 64)
#pragma unroll
    for (int nt = 0; nt < 4; ++nt) {
        int col = nt * 16 + n;
#pragma unroll
        for (int kc = 0; kc < 4; ++kc) {
            int k = kc * 4 + 2 * half;
            bx[nt][kc].x = wx[(k + 0) * V_ + col];
            bx[nt][kc].y = wx[(k + 1) * V_ + col];
        }
#pragma unroll
        for (int kc = 0; kc < 2; ++kc) {
            int k = kc * 4 + 2 * half;
            bz[nt][kc].x = wz[(k + 0) * V_ + col];
            bz[nt][kc].y = wz[(k + 1) * V_ + col];
        }
    }

    const int nstrips = (B_ * T_) / 16;
    const int wave    = blockIdx.x * (blockDim.x >> 5) + (threadIdx.x >> 5);
    const int wstride = gridDim.x * (blockDim.x >> 5);

    const size_t cm_off = (size_t)B_ * T_ * V_;   // out_cm = out + cm_off

    for (int strip = wave; strip < nstrips; strip += wstride) {
        // Wave-uniform scalar bases.
        const int s_row0 = __builtin_amdgcn_readfirstlane(strip * 16);
        const int b_lo   = s_row0 / T_;                 // SALU magic-div, 1x/strip
        const int t_lo   = s_row0 - b_lo * T_;
        const int b_hi   = (b_lo + 1 < B_) ? b_lo + 1 : b_lo;

        // --- A-operands: lane (m, m+16) hold row m; K = chunk*4 + j + 2h. ---
        const float* xr = x + (size_t)(s_row0 + n) * C_;
        const float* zr = z + (size_t)(s_row0 + n) * L_;
        v2f ax[4], az[2];
#pragma unroll
        for (int kc = 0; kc < 4; ++kc)
            ax[kc] = *(const v2f*)(xr + kc * 4 + 2 * half);
#pragma unroll
        for (int kc = 0; kc < 2; ++kc)
            az[kc] = *(const v2f*)(zr + kc * 4 + 2 * half);

#pragma unroll
        for (int nt = 0; nt < 4; ++nt) {
            const int col = nt * 16 + n;
            // Only two possible batch ids across the 16 rows of this strip;
            // select per C/D row and seed the WMMA accumulator with it.
            const float uA = uterm[(size_t)b_lo * V_ + col];
            const float uB = uterm[(size_t)b_hi * V_ + col];
            v8f acc;
#pragma unroll
            for (int r = 0; r < 8; ++r)
                acc[r] = ((t_lo + r + 8 * half) >= T_) ? uB : uA;

#pragma unroll
            for (int kc = 0; kc < 4; ++kc)
                acc = __builtin_amdgcn_wmma_f32_16x16x4_f32(
                    false, ax[kc], false, bx[nt][kc], (short)0, acc, false, false);
#pragma unroll
            for (int kc = 0; kc < 2; ++kc)
                acc = __builtin_amdgcn_wmma_f32_16x16x4_f32(
                    false, az[kc], false, bz[nt][kc], (short)0, acc, false, false);

            // --- Epilogue: single-med3 clip + link + store. ---
            float* pl = out + (size_t)s_row0 * V_ + (size_t)(half * 8 * V_ + col);
            float* pc = pl + cm_off;
#pragma unroll
            for (int r = 0; r < 8; ++r) {
                const float lp = __builtin_amdgcn_fmed3f(acc[r], -CLIPV, CLIPV);
                const float cm = (col < NBERN) ? (1.0f / (1.0f + __expf(-lp)))
                                               : __expf(lp);
                pl[r * V_] = lp;
                pc[r * V_] = cm;
            }
        }
    }
}

extern "C" void kernel_launch(void* const* d_in, const int* in_sizes, int n_in,
                              void* d_out, int out_size, void* d_ws, size_t ws_size,
                              hipStream_t stream)
{
    const float* eps  = (const float*)d_in[0];
    const float* u    = (const float*)d_in[1];
    const float* x    = (const float*)d_in[2];
    const float* A    = (const float*)d_in[3];
    const float* lvz1 = (const float*)d_in[4];
    const float* b0   = (const float*)d_in[5];
    const float* b1   = (const float*)d_in[6];
    const float* itc  = (const float*)d_in[7];
    const float* wz   = (const float*)d_in[8];
    const float* wx   = (const float*)d_in[9];
    const float* lvu  = (const float*)d_in[10];

    float* z     = (float*)d_ws;                    // B*T*L floats (26.2 MB)
    float* uterm = z + (size_t)B_ * T_ * L_;        // B*V floats (1 MB)

    uterm_kernel<<<(B_ * V_ + 255) / 256, 256, 0, stream>>>(u, itc, lvu, uterm);

    var1_recursion_kernel<<<(B_ + 127) / 128, 128, 0, stream>>>(
        eps, A, lvz1, b0, b1, z);

    // 640 blocks * 8 waves = 5120 waves; 51200 strips -> 10 strips/wave.
    glm_wmma_kernel<<<640, 256, 0, stream>>>(
        x, z, uterm, wx, wz, (float*)d_out);
}